// GrangerNet_23381801960066
// MI455X (gfx1250) — compile-verified
//
#include <hip/hip_runtime.h>

namespace {
constexpr int kLag     = 20;
constexpr int kEff     = 26;
constexpr int kHid     = 32;
constexpr int kIn      = 39;       // 19 eff lags + 20 exp lags; K-loop padded to 40
constexpr int kBatch   = 131072;
constexpr int kRows    = 64;       // batch rows per workgroup (4 row-subtiles of 16)
constexpr int kThreads = 256;      // 8 waves (wave32)
}

typedef __attribute__((ext_vector_type(2))) float v2f;
typedef __attribute__((ext_vector_type(8))) float v8f;

// x_cat[b, f, k] = k < 19 ? x_eff[b, k, f] : (k < 39 ? x_exp[b, k-19] : 0)
__device__ __forceinline__ float xcat_ld(const float* __restrict__ xeff,
                                         const float* __restrict__ xexp,
                                         int b, int k, int f) {
    float v = 0.0f;
    if (k < kLag - 1)
        v = xeff[(b * (kLag - 1) + k) * kEff + f];
    else if (k < kIn)
        v = xexp[b * kLag + (k - (kLag - 1))];
    return v;
}

__global__ void __launch_bounds__(kThreads)
granger_wmma_f32(const float* __restrict__ x_exp,
                 const float* __restrict__ x_eff,
                 const float* __restrict__ W1,
                 const float* __restrict__ b1,
                 const float* __restrict__ W2,
                 const float* __restrict__ b2,
                 float* __restrict__ out)
{
    __shared__ float s_pred[kRows * kEff];   // [row][f], matches out layout

    const int b0    = blockIdx.x * kRows;
    const int lane  = threadIdx.x & 31;
    const int wave  = threadIdx.x >> 5;      // 0..7
    const int nm    = lane & 15;             // M row (A) / N col (B) within tile
    const int kh    = (lane >> 4) << 1;      // K half offset: 0 (lanes 0-15), 2 (16-31)
    const int rbase = (lane >> 4) << 3;      // C/D rows 0-7 vs 8-15

    // Each wave owns effects {wave, wave+8, wave+16, wave+24<26}
    for (int f = wave; f < kEff; f += 8) {
        const float* __restrict__ w1f = W1 + f * kIn * kHid;

        // Preload all W1[f] B-fragments for both hidden halves: 40 VGPRs.
        // B 4x16 tile layout (mirrors A's K split): component j holds
        // K = 4*ks + kh + j, N = nm (+16 for upper hidden half).
        v2f bA[10], bB[10];
        #pragma unroll
        for (int ks = 0; ks < 10; ++ks) {
            const int k0 = ks * 4 + kh;
            bA[ks].x = (k0     < kIn) ? w1f[(k0    ) * kHid + nm     ] : 0.0f;
            bA[ks].y = (k0 + 1 < kIn) ? w1f[(k0 + 1) * kHid + nm     ] : 0.0f;
            bB[ks].x = (k0     < kIn) ? w1f[(k0    ) * kHid + nm + 16] : 0.0f;
            bB[ks].y = (k0 + 1 < kIn) ? w1f[(k0 + 1) * kHid + nm + 16] : 0.0f;
        }
        const float b1a = b1[f * kHid + nm];
        const float b1b = b1[f * kHid + nm + 16];
        const float w2a = W2[f * kHid + nm];
        const float w2b = W2[f * kHid + nm + 16];
        const float b2f = b2[f];

        #pragma unroll
        for (int rt = 0; rt < kRows / 16; ++rt) {
            const int brow = b0 + rt * 16 + nm;

            v8f acc0 = {};
            v8f acc1 = {};
            #pragma unroll
            for (int ks = 0; ks < 10; ++ks) {
                const int k0 = ks * 4 + kh;
                // A 16x4 fp32 tile: lanes 0-15 K={4ks,4ks+1}, lanes 16-31 K={4ks+2,4ks+3}
                v2f a;
                a.x = xcat_ld(x_eff, x_exp, brow, k0,     f);
                a.y = xcat_ld(x_eff, x_exp, brow, k0 + 1, f);
                acc0 = __builtin_amdgcn_wmma_f32_16x16x4_f32(
                    false, a, false, bA[ks], (short)0, acc0, false, false);
                acc1 = __builtin_amdgcn_wmma_f32_16x16x4_f32(
                    false, a, false, bB[ks], (short)0, acc1, false, false);
            }

            // Layer 2: ReLU then weighted sum over the 32 hidden units.
            // Lane owns hidden column nm (acc0) and nm+16 (acc1); VGPR r holds
            // row rbase+r. Reduce across the 16 lanes of each half.
            float t[8];
            #pragma unroll
            for (int r = 0; r < 8; ++r) {
                float h0 = fmaxf(acc0[r] + b1a, 0.0f);
                float h1 = fmaxf(acc1[r] + b1b, 0.0f);
                t[r] = h0 * w2a + h1 * w2b;
            }
            #pragma unroll
            for (int mask = 1; mask < 16; mask <<= 1) {
                #pragma unroll
                for (int r = 0; r < 8; ++r)
                    t[r] += __shfl_xor(t[r], mask, 32);
            }
            if (nm == 0) {  // lanes 0 (rows 0-7) and 16 (rows 8-15)
                #pragma unroll
                for (int r = 0; r < 8; ++r)
                    s_pred[(rt * 16 + rbase + r) * kEff + f] = t[r] + b2f;
            }
        }
    }

    __syncthreads();
    // Coalesced contiguous store: tile [64 rows][26 f] == out[b0*26 ..)
    const size_t obase = (size_t)b0 * kEff;
    for (int i = threadIdx.x; i < kRows * kEff; i += kThreads)
        out[obase + i] = s_pred[i];
}

extern "C" void kernel_launch(void* const* d_in, const int* in_sizes, int n_in,
                              void* d_out, int out_size, void* d_ws, size_t ws_size,
                              hipStream_t stream) {
    const float* x_exp = (const float*)d_in[0];   // [B, 20, 1]
    const float* x_eff = (const float*)d_in[1];   // [B, 19, 26]
    const float* W1    = (const float*)d_in[2];   // [26, 39, 32]
    const float* b1    = (const float*)d_in[3];   // [26, 32]
    const float* W2    = (const float*)d_in[4];   // [26, 32]
    const float* b2    = (const float*)d_in[5];   // [26]
    float* out = (float*)d_out;                   // [B, 26]

    dim3 grid(kBatch / kRows);                    // 2048 blocks
    granger_wmma_f32<<<grid, dim3(kThreads), 0, stream>>>(
        x_exp, x_eff, W1, b1, W2, b2, out);
}